// GrenolNet_73100343378646
// MI455X (gfx1250) — compile-verified
//
#include <hip/hip_runtime.h>
#include <math.h>

typedef __attribute__((ext_vector_type(2))) float v2f;
typedef __attribute__((ext_vector_type(8))) float v8f;
typedef __attribute__((ext_vector_type(4))) unsigned int u32x4;
typedef __attribute__((ext_vector_type(8))) int i32x8;
typedef __attribute__((ext_vector_type(4))) int i32x4;

#define BB 8
#define NN 128
#define FF 8

// D = A(16x4) x B(4x16) + C chained twice for K=8, all fp32.
__device__ inline v8f wmma_k8(v2f a1, v2f b1, v2f a2, v2f b2, v8f c) {
  c = __builtin_amdgcn_wmma_f32_16x16x4_f32(false, a1, false, b1, (short)0, c, false, false);
  c = __builtin_amdgcn_wmma_f32_16x16x4_f32(false, a2, false, b2, (short)0, c, false, false);
  return c;
}

// Stage an (8 x NCOL) f32 row-major matrix into LDS.
// Preferred path: Tensor Data Mover (async DMA, tracked by TENSORcnt).
// Fallback: manual cooperative copy.
template<int NCOL>
__device__ inline void stage_weights(float* ldsDst, const float* gsrc, int tid) {
#if __has_builtin(__builtin_amdgcn_tensor_load_to_lds)
  if (tid < 32) {
    const unsigned long long ga = (unsigned long long)(uintptr_t)gsrc;
    u32x4 g0;
    g0[0] = 1u;                                        // count=1 (valid), user mode
    g0[1] = (unsigned int)(uintptr_t)ldsDst;           // lds_addr (low 32 bits = LDS offset)
    g0[2] = (unsigned int)ga;                          // global_addr[31:0]
    g0[3] = ((unsigned int)(ga >> 32) & 0x1FFFFFFu)    // global_addr[56:32]
            | (2u << 30);                              // type = 2 ("image")
    i32x8 g1;
    g1[0] = (int)(2u << 16);                           // data_size = 4 bytes (code 2)
    g1[1] = (int)(((unsigned)NCOL & 0xFFFFu) << 16);   // tensor_dim0 low16 (bits 63:48)
    g1[2] = (int)((((unsigned)NCOL >> 16) & 0xFFFFu)   // tensor_dim0 hi16 (bits 79:64)
            | (8u << 16));                             // tensor_dim1 = 8  (bits 95:80)
    g1[3] = (int)(((unsigned)NCOL) << 16);             // tile_dim0 = NCOL (bits 127:112)
    g1[4] = 8;                                         // tile_dim1 = 8
    g1[5] = (int)NCOL;                                 // tensor_dim0_stride low32
    g1[6] = 0;
    g1[7] = 0;
    i32x4 z4; z4[0] = 0; z4[1] = 0; z4[2] = 0; z4[3] = 0;
    i32x8 z8;
    #pragma unroll
    for (int q = 0; q < 8; ++q) z8[q] = 0;
    __builtin_amdgcn_tensor_load_to_lds(g0, g1, z4, z4, z8, 0);
    __builtin_amdgcn_s_wait_tensorcnt(0);
  }
#else
  for (int idx = tid; idx < NCOL * 8; idx += 128) ldsDst[idx] = gsrc[idx];
#endif
}

// One workgroup = (batch b, 16 destination nodes). 4 waves split the 128 sources.
// T = relu(ea_tile @ we + be)  via WMMA f32 (K=8 -> 2 chained 16x16x4)
// msg[i,o] += x[b,j,in] * T[i,(in,o)]  (scalar fold, src row uniform over tile)
// out = msg/128 + x @ wr + br
template<int IN_C, int OUT_C>
__global__ __launch_bounds__(128) void nnconv_kernel(
    const float* __restrict__ x,   // (BN, IN_C)
    const float* __restrict__ ea,  // (E, 8)
    const float* __restrict__ we,  // (8, IN_C*OUT_C)
    const float* __restrict__ be,  // (IN_C*OUT_C)
    const float* __restrict__ wr,  // (IN_C, OUT_C)
    const float* __restrict__ br,  // (OUT_C)
    float* __restrict__ out)       // (BN, OUT_C)
{
  constexpr int NCOL = IN_C * OUT_C;
  constexpr int NT   = NCOL / 16;                    // 16-column tiles
  constexpr int NA   = (OUT_C >= 16) ? (OUT_C / 16) : 1;

  __shared__ float weS[NCOL * 8];                    // row-major [k][col] (TDM-friendly)
  __shared__ float beS[NCOL];
  __shared__ float red[4 * NA * 32 * 8];             // cross-wave reduction

  const int tid = threadIdx.x;
  stage_weights<NCOL>(weS, we, tid);
  for (int idx = tid; idx < NCOL; idx += 128) beS[idx] = be[idx];
  __syncthreads();

  const int wave = tid >> 5;
  const int lane = tid & 31;
  const int col  = lane & 15;
  const int k0   = (lane < 16) ? 0 : 2;
  const int b    = blockIdx.x >> 3;                  // 8 dst tiles per batch
  const int i0   = (blockIdx.x & 7) * 16;

  v8f acc[NA];
  #pragma unroll
  for (int h = 0; h < NA; ++h)
    #pragma unroll
    for (int e = 0; e < 8; ++e) acc[h][e] = 0.0f;

  const int jbeg = wave * 32, jend = jbeg + 32;
  for (int j = jbeg; j < jend; ++j) {
    // A fragment: 16 edges (dst i0..i0+15, source j), K = 8 edge features
    const float* arow = ea + ((size_t)((b * NN + j) * NN) + i0 + col) * FF;
    __builtin_prefetch(arow + NN * FF, 0, 1);        // next j's row -> global_prefetch
    v2f a1, a2;
    a1.x = arow[k0];     a1.y = arow[k0 + 1];
    a2.x = arow[k0 + 4]; a2.y = arow[k0 + 5];
    // source-row features, one per lane, broadcast later via shuffle
    float xj = (lane < IN_C) ? x[(b * NN + j) * IN_C + lane] : 0.0f;

    for (int t = 0; t < NT; ++t) {
      const int c = t * 16 + col;
      v2f b1, b2;
      b1.x = weS[k0 * NCOL + c];       b1.y = weS[(k0 + 1) * NCOL + c];
      b2.x = weS[(k0 + 4) * NCOL + c]; b2.y = weS[(k0 + 5) * NCOL + c];
      const float bias = beS[c];
      v8f tv;
      #pragma unroll
      for (int e = 0; e < 8; ++e) tv[e] = bias;
      tv = wmma_k8(a1, b1, a2, b2, tv);
      const int in   = (t * 16 + col) / OUT_C;       // input channel of this column
      const float mult = __shfl(xj, in, 32);
      const int ah = (OUT_C >= 16) ? (t % NA) : 0;
      #pragma unroll
      for (int e = 0; e < 8; ++e)
        acc[ah][e] += mult * fmaxf(tv[e], 0.0f);     // relu then fold
    }
  }

  // reduce the 4 waves' partial sums through LDS
  #pragma unroll
  for (int h = 0; h < NA; ++h)
    #pragma unroll
    for (int e = 0; e < 8; ++e)
      red[((wave * NA + h) * 32 + lane) * 8 + e] = acc[h][e];
  __syncthreads();

  if (wave == 0) {
    #pragma unroll
    for (int h = 0; h < NA; ++h)
      #pragma unroll
      for (int e = 0; e < 8; ++e) {
        float s = 0.0f;
        for (int w = 0; w < 4; ++w)
          s += red[((w * NA + h) * 32 + lane) * 8 + e];
        acc[h][e] = s;
      }
    if (OUT_C < 16) {                                // fold duplicated column halves
      #pragma unroll
      for (int e = 0; e < 8; ++e)
        acc[0][e] += __shfl_xor(acc[0][e], 8, 32);
    }
    #pragma unroll
    for (int h = 0; h < NA; ++h) {
      const int o = (OUT_C >= 16) ? (h * 16 + col) : col;
      const bool valid = (OUT_C >= 16) || (col < OUT_C);
      if (valid) {
        #pragma unroll
        for (int v = 0; v < 8; ++v) {
          const int row = i0 + v + ((lane >= 16) ? 8 : 0);
          const float* xr = x + (b * NN + row) * IN_C;
          float root = br[o];
          #pragma unroll
          for (int k = 0; k < IN_C; ++k) root += xr[k] * wr[k * OUT_C + o];
          out[(b * NN + row) * OUT_C + o] = acc[h][v] * (1.0f / 128.0f) + root;
        }
      }
    }
  }
}

// sinusoidal time embedding + 2-layer MLP with exact GELU -> te (B, N)
__global__ __launch_bounds__(256) void time_kernel(
    const float* __restrict__ ts,
    const float* __restrict__ wt1, const float* __restrict__ bt1,
    const float* __restrict__ wt2, const float* __restrict__ bt2,
    float* __restrict__ te)
{
  __shared__ float emb[BB * NN];
  __shared__ float t1[BB * NN];
  const int tid = threadIdx.x;
  const float lf = logf(10000.0f) / 63.0f;
  for (int idx = tid; idx < BB * NN; idx += 256) {
    int bb = idx / NN, k = idx % NN;
    float t = ts[bb];
    float v;
    if (k < 64) { float fr = expf(-(float)k * lf);        v = sinf(t * fr); }
    else        { float fr = expf(-(float)(k - 64) * lf); v = cosf(t * fr); }
    emb[idx] = v;
  }
  __syncthreads();
  for (int idx = tid; idx < BB * NN; idx += 256) {
    int bb = idx / NN, n = idx % NN;
    float s = bt1[n];
    for (int k = 0; k < NN; ++k) s += emb[bb * NN + k] * wt1[k * NN + n];
    t1[idx] = 0.5f * s * (1.0f + erff(s * 0.70710678118654752f));
  }
  __syncthreads();
  for (int idx = tid; idx < BB * NN; idx += 256) {
    int bb = idx / NN, n = idx % NN;
    float s = bt2[n];
    for (int k = 0; k < NN; ++k) s += t1[bb * NN + k] * wt2[k * NN + n];
    te[idx] = s;
  }
}

// BatchNorm1d(n_nodes) batch stats: per node-index n over (B, F) = 64 values
__global__ __launch_bounds__(128) void bn_kernel(
    const float* __restrict__ xn, const float* __restrict__ gamma,
    float* __restrict__ mean_out, float* __restrict__ rsg_out)
{
  const int n = threadIdx.x;
  float s = 0.0f, s2 = 0.0f;
  for (int b = 0; b < BB; ++b)
    #pragma unroll
    for (int f = 0; f < FF; ++f) {
      float v = xn[(b * NN + n) * FF + f];
      s += v; s2 += v * v;
    }
  float m = s * (1.0f / 64.0f);
  float var = s2 * (1.0f / 64.0f) - m * m;
  mean_out[n] = m;
  rsg_out[n]  = rsqrtf(var + 1e-5f) * gamma[n];
}

// fc_mapping (8->128 relu ->128 relu ->8 sigmoid) via WMMA, fused with
// te scaling, BatchNorm and final subtraction. 1 wave per 16-node row tile.
__global__ __launch_bounds__(32) void final_kernel(
    const float* __restrict__ h3,
    const float* __restrict__ wf1, const float* __restrict__ bf1,
    const float* __restrict__ wf2, const float* __restrict__ bf2,
    const float* __restrict__ wf3, const float* __restrict__ bf3,
    const float* __restrict__ te, const float* __restrict__ xnoisy,
    const float* __restrict__ meanv, const float* __restrict__ rsg,
    const float* __restrict__ beta, float* __restrict__ out)
{
  __shared__ float m1[16 * 128];
  __shared__ float m2[16 * 128];
  const int lane = threadIdx.x;
  const int col  = lane & 15;
  const int k0   = (lane < 16) ? 0 : 2;
  const int r0   = blockIdx.x * 16;
  const int Moff = (lane >= 16) ? 8 : 0;

  // GEMM1: m1 = relu(h3(16x8) @ wf1(8x128) + bf1), K=8
  {
    const float* ar = h3 + (r0 + col) * FF;
    v2f a1, a2;
    a1.x = ar[k0];     a1.y = ar[k0 + 1];
    a2.x = ar[k0 + 4]; a2.y = ar[k0 + 5];
    for (int nt = 0; nt < 8; ++nt) {
      const int nc = nt * 16 + col;
      v2f b1, b2;
      b1.x = wf1[k0 * 128 + nc];       b1.y = wf1[(k0 + 1) * 128 + nc];
      b2.x = wf1[(k0 + 4) * 128 + nc]; b2.y = wf1[(k0 + 5) * 128 + nc];
      v8f c; const float bias = bf1[nc];
      #pragma unroll
      for (int e = 0; e < 8; ++e) c[e] = bias;
      c = wmma_k8(a1, b1, a2, b2, c);
      #pragma unroll
      for (int v = 0; v < 8; ++v)
        m1[(v + Moff) * 128 + nc] = fmaxf(c[v], 0.0f);
    }
  }
  __syncthreads();

  // GEMM2: m2 = relu(m1(16x128) @ wf2(128x128) + bf2), K=128 -> 32 chunks
  for (int nt = 0; nt < 8; ++nt) {
    const int nc = nt * 16 + col;
    v8f c; const float bias = bf2[nc];
    #pragma unroll
    for (int e = 0; e < 8; ++e) c[e] = bias;
    for (int q = 0; q < 32; ++q) {
      const int kk = q * 4 + k0;
      v2f aa, bb;
      aa.x = m1[col * 128 + kk];   aa.y = m1[col * 128 + kk + 1];
      bb.x = wf2[kk * 128 + nc];   bb.y = wf2[(kk + 1) * 128 + nc];
      c = __builtin_amdgcn_wmma_f32_16x16x4_f32(false, aa, false, bb, (short)0, c, false, false);
    }
    #pragma unroll
    for (int v = 0; v < 8; ++v)
      m2[(v + Moff) * 128 + nc] = fmaxf(c[v], 0.0f);
  }
  __syncthreads();

  // GEMM3: m3 = sigmoid(m2(16x128) @ wf3(128x8) + bf3); combine with bn, te
  {
    v8f c; const float bias = (col < 8) ? bf3[col] : 0.0f;
    #pragma unroll
    for (int e = 0; e < 8; ++e) c[e] = bias;
    for (int q = 0; q < 32; ++q) {
      const int kk = q * 4 + k0;
      v2f aa, bb;
      aa.x = m2[col * 128 + kk]; aa.y = m2[col * 128 + kk + 1];
      if (col < 8) { bb.x = wf3[kk * 8 + col]; bb.y = wf3[(kk + 1) * 8 + col]; }
      else         { bb.x = 0.0f;              bb.y = 0.0f; }
      c = __builtin_amdgcn_wmma_f32_16x16x4_f32(false, aa, false, bb, (short)0, c, false, false);
    }
    if (col < 8) {
      #pragma unroll
      for (int v = 0; v < 8; ++v) {
        const int node = r0 + v + Moff;
        const int bb_  = node >> 7;
        const int n    = node & 127;
        const float m3 = 1.0f / (1.0f + expf(-c[v]));
        const float mapped = m3 * te[bb_ * NN + n];
        const float xv = xnoisy[node * FF + col];
        const float bn = (xv - meanv[n]) * rsg[n] + beta[n];
        out[node * FF + col] = bn - mapped;
      }
    }
  }
}

extern "C" void kernel_launch(void* const* d_in, const int* in_sizes, int n_in,
                              void* d_out, int out_size, void* d_ws, size_t ws_size,
                              hipStream_t stream) {
  (void)in_sizes; (void)n_in; (void)out_size; (void)ws_size;
  const float* x_noisy = (const float*)d_in[0];
  const float* x_src   = (const float*)d_in[1];
  const float* ea      = (const float*)d_in[2];
  const float* ts      = (const float*)d_in[3];
  const float* we1 = (const float*)d_in[4];  const float* be1 = (const float*)d_in[5];
  const float* wr1 = (const float*)d_in[6];  const float* br1 = (const float*)d_in[7];
  const float* we2 = (const float*)d_in[8];  const float* be2 = (const float*)d_in[9];
  const float* wr2 = (const float*)d_in[10]; const float* br2 = (const float*)d_in[11];
  const float* we3 = (const float*)d_in[12]; const float* be3 = (const float*)d_in[13];
  const float* wr3 = (const float*)d_in[14]; const float* br3 = (const float*)d_in[15];
  const float* wt1 = (const float*)d_in[16]; const float* bt1 = (const float*)d_in[17];
  const float* wt2 = (const float*)d_in[18]; const float* bt2 = (const float*)d_in[19];
  const float* wf1 = (const float*)d_in[20]; const float* bf1 = (const float*)d_in[21];
  const float* wf2 = (const float*)d_in[22]; const float* bf2 = (const float*)d_in[23];
  const float* wf3 = (const float*)d_in[24]; const float* bf3 = (const float*)d_in[25];
  const float* gamma = (const float*)d_in[26];
  const float* beta  = (const float*)d_in[27];
  float* out = (float*)d_out;

  float* ws   = (float*)d_ws;
  float* h1   = ws;            // 1024*32
  float* h2   = ws + 32768;    // 1024*32
  float* h3   = ws + 65536;    // 1024*8
  float* te   = ws + 73728;    // 8*128
  float* mnv  = ws + 74752;    // 128
  float* rsg  = ws + 74880;    // 128

  nnconv_kernel<8, 32><<<64, 128, 0, stream>>>(x_src, ea, we1, be1, wr1, br1, h1);
  nnconv_kernel<32, 32><<<64, 128, 0, stream>>>(h1, ea, we2, be2, wr2, br2, h2);
  nnconv_kernel<32, 8><<<64, 128, 0, stream>>>(h2, ea, we3, be3, wr3, br3, h3);
  time_kernel<<<1, 256, 0, stream>>>(ts, wt1, bt1, wt2, bt2, te);
  bn_kernel<<<1, 128, 0, stream>>>(x_noisy, gamma, mnv, rsg);
  final_kernel<<<64, 32, 0, stream>>>(h3, wf1, bf1, wf2, bf2, wf3, bf3,
                                      te, x_noisy, mnv, rsg, beta, out);
}